// PhraseAttention_8933531976139
// MI455X (gfx1250) — compile-verified
//
#include <hip/hip_runtime.h>
#include <hip/hip_bf16.h>
#include <math.h>

typedef __attribute__((ext_vector_type(2))) float v2f;
typedef __attribute__((ext_vector_type(8))) float v8f;

// ---------------------------------------------------------------------------
// Kernel 1: scores[b,s] = dot(qry[b,:], ctx[b,s,:]) + log(mask[b,s])
// One wave32 computes 16 consecutive s-rows with V_WMMA_F32_16X16X4_F32.
// A tile (16x4 f32): lanes 0-15 carry K=0,1 (VGPR0,1); lanes 16-31 carry K=2,3.
// B tile (4x16 f32): qry replicated across all 16 columns -> every column of D
// holds the same dot product; lane 0 ends with M=0..7, lane 16 with M=8..15.
// ---------------------------------------------------------------------------
__global__ void pa_scores_wmma_kernel(const float* __restrict__ qry,
                                      const float* __restrict__ ctx,
                                      const float* __restrict__ mask,
                                      float* __restrict__ scores,
                                      int B, int S, int D, int tilesPerB) {
    const int b    = blockIdx.x / tilesPerB;
    const int tile = blockIdx.x % tilesPerB;
    const int wave = threadIdx.x >> 5;
    const int lane = threadIdx.x & 31;
    const int s0   = tile * 64 + wave * 16;          // 4 waves x 16 rows / block

    if (s0 + 16 <= S) {
        // ---- full 16-row tile: WMMA path (wave-uniform branch, EXEC all-1) ----
        const int row  = lane & 15;                  // M
        const int kOff = (lane >> 4) ? 2 : 0;        // this half-wave's K pair
        const float* __restrict__ aRow = ctx + ((size_t)b * S + (s0 + row)) * D + kOff;
        const float* __restrict__ q    = qry + (size_t)b * D + kOff;

        v8f acc0 = {};
        v8f acc1 = {};
        const int D8 = D & ~7;
        int d = 0;
        for (; d < D8; d += 8) {
            v2f a0, b0, a1, b1;
            a0[0] = aRow[d];     a0[1] = aRow[d + 1];
            b0[0] = q[d];        b0[1] = q[d + 1];
            a1[0] = aRow[d + 4]; a1[1] = aRow[d + 5];
            b1[0] = q[d + 4];    b1[1] = q[d + 5];
            acc0 = __builtin_amdgcn_wmma_f32_16x16x4_f32(
                false, a0, false, b0, (short)0, acc0, false, false);
            acc1 = __builtin_amdgcn_wmma_f32_16x16x4_f32(
                false, a1, false, b1, (short)0, acc1, false, false);
        }
        if (d + 4 <= D) {                            // one leftover K=4 step
            v2f a0, b0;
            a0[0] = aRow[d]; a0[1] = aRow[d + 1];
            b0[0] = q[d];    b0[1] = q[d + 1];
            acc0 = __builtin_amdgcn_wmma_f32_16x16x4_f32(
                false, a0, false, b0, (short)0, acc0, false, false);
            d += 4;
        }
        const int dTail = d;                         // D % 4 handled scalar below
        v8f acc = acc0 + acc1;

        if (lane == 0 || lane == 16) {
            const int mbase = (lane == 16) ? 8 : 0;
            for (int r = 0; r < 8; ++r) {
                const int s = s0 + mbase + r;
                float v = acc[r];
                for (int dd = dTail; dd < D; ++dd)   // generic tail (no-op here)
                    v += ctx[((size_t)b * S + s) * D + dd] * qry[(size_t)b * D + dd];
                scores[(size_t)b * S + s] = v + logf(mask[(size_t)b * S + s]);
            }
        }
    } else {
        // ---- ragged tail tile: plain per-lane dot products ----
        const int s = s0 + (lane & 15);
        if (lane < 16 && s < S) {
            const float* __restrict__ cr = ctx + ((size_t)b * S + s) * D;
            const float* __restrict__ q  = qry + (size_t)b * D;
            float v = 0.f;
            for (int dd = 0; dd < D; ++dd) v += cr[dd] * q[dd];
            scores[(size_t)b * S + s] = v + logf(mask[(size_t)b * S + s]);
        }
    }
}

// ---------------------------------------------------------------------------
// Kernel 2: softmax over S, gather parent row, alphas = min(parent, softmax)
// One block (256 threads) per batch element.
// ---------------------------------------------------------------------------
__global__ void pa_softmax_min_kernel(const float* __restrict__ scores,
                                      const float* __restrict__ prevatts,
                                      const int* __restrict__ parent_ptr,
                                      float* __restrict__ alphas,
                                      int S, int T) {
    const int b   = blockIdx.x;
    const int tid = threadIdx.x;
    __shared__ float red[256];
    const float* __restrict__ sc = scores + (size_t)b * S;

    float lmax = -INFINITY;
    for (int s = tid; s < S; s += 256) lmax = fmaxf(lmax, sc[s]);
    red[tid] = lmax;
    __syncthreads();
    for (int off = 128; off > 0; off >>= 1) {
        if (tid < off) red[tid] = fmaxf(red[tid], red[tid + off]);
        __syncthreads();
    }
    const float m = red[0];
    __syncthreads();

    float lsum = 0.f;
    for (int s = tid; s < S; s += 256) lsum += expf(sc[s] - m);
    red[tid] = lsum;
    __syncthreads();
    for (int off = 128; off > 0; off >>= 1) {
        if (tid < off) red[tid] += red[tid + off];
        __syncthreads();
    }
    const float inv = 1.0f / red[0];

    const float* __restrict__ par =
        prevatts + ((size_t)b * T + parent_ptr[b]) * S;
    for (int s = tid; s < S; s += 256) {
        const float a = expf(sc[s] - m) * inv;
        alphas[(size_t)b * S + s] = fminf(par[s], a);
    }
}

// ---------------------------------------------------------------------------
// Kernel 3: summary[b,d] = sum_s alphas[b,s] * ctx[b,s,d]
// Block = 256 consecutive d-columns of one b; alpha chunk staged in LDS;
// ctx reads are fully coalesced (this pass sets the 23.3 TB/s HBM floor).
// ---------------------------------------------------------------------------
__global__ void pa_summary_kernel(const float* __restrict__ alphas,
                                  const float* __restrict__ ctx,
                                  float* __restrict__ summary,
                                  int S, int D) {
    const int b   = blockIdx.y;
    const int tid = threadIdx.x;
    const int d   = blockIdx.x * 256 + tid;
    __shared__ float sal[256];

    float acc = 0.f;
    for (int s0 = 0; s0 < S; s0 += 256) {
        const int sidx = s0 + tid;
        sal[tid] = (sidx < S) ? alphas[(size_t)b * S + sidx] : 0.f;
        __syncthreads();
        const int lim = (S - s0 < 256) ? (S - s0) : 256;
        if (d < D) {
            const float* __restrict__ cp = ctx + ((size_t)b * S + s0) * D + d;
            for (int j = 0; j < lim; ++j) acc += sal[j] * cp[(size_t)j * D];
        }
        __syncthreads();
    }
    if (d < D) summary[(size_t)b * D + d] = acc;
}

// ---------------------------------------------------------------------------
extern "C" void kernel_launch(void* const* d_in, const int* in_sizes, int n_in,
                              void* d_out, int out_size, void* d_ws, size_t ws_size,
                              hipStream_t stream) {
    const float* qry      = (const float*)d_in[0];   // (B, D)
    const float* ctx      = (const float*)d_in[1];   // (B, S, D)
    const float* mask     = (const float*)d_in[2];   // (B, S)
    const float* prevatts = (const float*)d_in[3];   // (B, T, S)
    const int*   parent   = (const int*)d_in[4];     // (B,)

    const int B = in_sizes[4];
    const int S = in_sizes[2] / B;
    const int D = in_sizes[0] / B;
    const int T = in_sizes[3] / (B * S);

    float* alphas  = (float*)d_out;                  // (B, S)
    float* summary = alphas + (size_t)B * S;         // (B, D)
    float* scores  = summary + (size_t)B * D;        // (B, S)

    // Kernel 1: 128 threads = 4 wave32s, each wave -> 16 s-rows.
    const int tilesPerB = (S + 63) / 64;
    pa_scores_wmma_kernel<<<dim3(B * tilesPerB), dim3(128), 0, stream>>>(
        qry, ctx, mask, scores, B, S, D, tilesPerB);

    // Kernel 2: one block per batch element.
    pa_softmax_min_kernel<<<dim3(B), dim3(256), 0, stream>>>(
        scores, prevatts, parent, alphas, S, T);

    // Kernel 3: (D/256) x B blocks.
    pa_summary_kernel<<<dim3((D + 255) / 256, B), dim3(256), 0, stream>>>(
        alphas, ctx, summary, S, D);
}